// SWEGNNProcessor_21620865368572
// MI455X (gfx1250) — compile-verified
//
#include <hip/hip_runtime.h>
#include <math.h>

typedef __attribute__((ext_vector_type(2))) float v2f;
typedef __attribute__((ext_vector_type(8))) float v8f;

#define WAVE_SZ 32
#define WAVES_PER_BLOCK 8
#define BLOCK (WAVE_SZ * WAVES_PER_BLOCK)      // 256 threads = 8 wave32
#define EDGES_PER_WAVE 16                      // one 16-row WMMA tile per wave
#define EDGES_PER_BLOCK (WAVES_PER_BLOCK * EDGES_PER_WAVE)

#define S_FEAT 5
#define D_FEAT 2
#define E_FEAT 3
#define HIDDEN 4
#define MLP_IN 17          // E_FEAT + 2*S_FEAT + 2*D_FEAT
#define APAD 20            // K padded to 5 chunks of 4 (zero fill)

// out[n] = x_t[n] @ filt[0];  acc = 0
__global__ __launch_bounds__(BLOCK) void swegnn_init_kernel(
    const float* __restrict__ x_t, const float* __restrict__ filt0,
    float* __restrict__ out, float* __restrict__ acc, int n_nodes)
{
    int i = blockIdx.x * blockDim.x + threadIdx.x;
    if (i < n_nodes) {
        float a0 = x_t[i * 2 + 0], a1 = x_t[i * 2 + 1];
        out[i * 2 + 0] = a0 * filt0[0] + a1 * filt0[2];
        out[i * 2 + 1] = a0 * filt0[1] + a1 * filt0[3];
        acc[i * 2 + 0] = 0.0f;
        acc[i * 2 + 1] = 0.0f;
    }
}

// out[n] += acc[n] @ filt_k;  acc = 0 (ready for next step)
__global__ __launch_bounds__(BLOCK) void swegnn_node_kernel(
    float* __restrict__ out, float* __restrict__ acc,
    const float* __restrict__ filtk, int n_nodes)
{
    int i = blockIdx.x * blockDim.x + threadIdx.x;
    if (i < n_nodes) {
        float a0 = acc[i * 2 + 0], a1 = acc[i * 2 + 1];
        out[i * 2 + 0] += a0 * filtk[0] + a1 * filtk[2];
        out[i * 2 + 1] += a0 * filtk[1] + a1 * filtk[3];
        acc[i * 2 + 0] = 0.0f;
        acc[i * 2 + 1] = 0.0f;
    }
}

// Per edge: gather 17-feature vector, MLP(17->4 PReLU ->2) via f32 WMMA,
// L2-normalize, mask, atomic scatter-add of (out[to]-out[frm])*w into acc[to].
__global__ __launch_bounds__(BLOCK) void swegnn_edge_kernel(
    const float* __restrict__ x_s, const float* __restrict__ out,
    const float* __restrict__ edge_attr,
    const float* __restrict__ w1, const float* __restrict__ b1,
    const float* __restrict__ alpha_p,
    const float* __restrict__ w2, const float* __restrict__ b2,
    const int* __restrict__ frm, const int* __restrict__ to,
    float* __restrict__ acc, int n_edges)
{
    __shared__ float sE[WAVES_PER_BLOCK][EDGES_PER_WAVE][APAD];    // feature tile
    __shared__ float sH[WAVES_PER_BLOCK][EDGES_PER_WAVE][HIDDEN];  // layer-1 out
    __shared__ float sW[WAVES_PER_BLOCK][EDGES_PER_WAVE][D_FEAT];  // layer-2 out

    const int tid   = threadIdx.x;
    const int wave  = tid / WAVE_SZ;
    const int lane  = tid % WAVE_SZ;
    const int m     = lane & 15;        // tile row (edge) / B,C column index
    const int khalf = lane >> 4;        // 0: K pair {0,1}; 1: K pair {2,3}
    const float alpha = alpha_p[0];

    // ---- preload B fragments (weights padded to 16 columns with zeros) ----
    v2f bw1[5];
#pragma unroll
    for (int c = 0; c < 5; ++c) {
        int k0 = 4 * c + 2 * khalf;
        float v0 = 0.0f, v1 = 0.0f;
        if (m < HIDDEN) {
            if (k0     < MLP_IN) v0 = w1[k0 * HIDDEN + m];
            if (k0 + 1 < MLP_IN) v1 = w1[(k0 + 1) * HIDDEN + m];
        }
        bw1[c][0] = v0; bw1[c][1] = v1;
    }
    v2f bw2;
    {
        int k0 = 2 * khalf;
        bw2[0] = (m < D_FEAT) ? w2[k0 * D_FEAT + m]       : 0.0f;
        bw2[1] = (m < D_FEAT) ? w2[(k0 + 1) * D_FEAT + m] : 0.0f;
    }
    const float c1init = (m < HIDDEN) ? b1[m] : 0.0f;   // bias folded into C
    const float c2init = (m < D_FEAT) ? b2[m] : 0.0f;

    const int edge0 = blockIdx.x * EDGES_PER_BLOCK + wave * EDGES_PER_WAVE;
    const int e     = edge0 + m;
    const bool valid = (e < n_edges);
    const int ec    = valid ? e : 0;

    // ---- gather: lanes 0-15 own an edge; lanes 16-31 fetch the other half ----
    int   tnode = 0;
    float of0 = 0.f, of1 = 0.f, ot0 = 0.f, ot1 = 0.f;
    float* row = sE[wave][m];
    if (lane < EDGES_PER_WAVE) {
        int f = frm[ec];
        int t = to[ec];
        tnode = t;
#pragma unroll
        for (int j = 0; j < S_FEAT; ++j) row[j] = x_s[f * S_FEAT + j];
        of0 = out[f * 2 + 0]; of1 = out[f * 2 + 1];
        ot0 = out[t * 2 + 0]; ot1 = out[t * 2 + 1];
        row[10] = of0; row[11] = of1; row[12] = ot0; row[13] = ot1;
    } else {
        int t = to[ec];
#pragma unroll
        for (int j = 0; j < S_FEAT; ++j) row[S_FEAT + j] = x_s[t * S_FEAT + j];
#pragma unroll
        for (int j = 0; j < E_FEAT; ++j) row[14 + j] = edge_attr[ec * E_FEAT + j];
        row[17] = 0.0f; row[18] = 0.0f; row[19] = 0.0f;   // K padding
    }
    __syncthreads();

    // ---- layer 1: h = e_ij @ w1 + b1  (5 chained f32 WMMAs, K=4 each) ----
    v8f h = { c1init, c1init, c1init, c1init, c1init, c1init, c1init, c1init };
#pragma unroll
    for (int c = 0; c < 5; ++c) {
        v2f a;
        a[0] = sE[wave][m][4 * c + 2 * khalf + 0];
        a[1] = sE[wave][m][4 * c + 2 * khalf + 1];
        h = __builtin_amdgcn_wmma_f32_16x16x4_f32(
                false, a, false, bw1[c], (short)0, h, false, false);
    }
    // PReLU elementwise in C/D layout
#pragma unroll
    for (int r = 0; r < 8; ++r) h[r] = (h[r] >= 0.0f) ? h[r] : alpha * h[r];

    // re-layout h (C/D layout -> A fragment) through LDS
    if (m < HIDDEN) {
#pragma unroll
        for (int r = 0; r < 8; ++r) sH[wave][r + 8 * khalf][m] = h[r];
    }
    __syncthreads();

    // ---- layer 2: w = h @ w2 + b2  (one f32 WMMA, K=4) ----
    v8f wv = { c2init, c2init, c2init, c2init, c2init, c2init, c2init, c2init };
    {
        v2f a2;
        a2[0] = sH[wave][m][2 * khalf + 0];
        a2[1] = sH[wave][m][2 * khalf + 1];
        wv = __builtin_amdgcn_wmma_f32_16x16x4_f32(
                 false, a2, false, bw2, (short)0, wv, false, false);
    }
    if (m < D_FEAT) {
#pragma unroll
        for (int r = 0; r < 8; ++r) sW[wave][r + 8 * khalf][m] = wv[r];
    }
    __syncthreads();

    // ---- epilogue: normalize, NaN->0, edge mask, scatter-add (lanes 0-15) ----
    if (lane < EDGES_PER_WAVE && valid) {
        float w0 = sW[wave][lane][0];
        float w1v = sW[wave][lane][1];
        float nrm = sqrtf(w0 * w0 + w1v * w1v);
        float q0 = w0 / nrm;
        float q1 = w1v / nrm;
        q0 = (q0 == q0) ? q0 : 0.0f;   // isnan -> 0
        q1 = (q1 == q1) ? q1 : 0.0f;
        bool emask = ((of0 + of1) != 0.0f) || ((ot0 + ot1) != 0.0f);
        if (emask) {
            atomicAdd(&acc[tnode * 2 + 0], (ot0 - of0) * q0);
            atomicAdd(&acc[tnode * 2 + 1], (ot1 - of1) * q1);
        }
    }
}

extern "C" void kernel_launch(void* const* d_in, const int* in_sizes, int n_in,
                              void* d_out, int out_size, void* d_ws, size_t ws_size,
                              hipStream_t stream)
{
    const float* x_s       = (const float*)d_in[0];
    const float* x_t       = (const float*)d_in[1];
    const float* edge_attr = (const float*)d_in[2];
    const float* filt      = (const float*)d_in[3];
    const float* w1        = (const float*)d_in[4];
    const float* b1        = (const float*)d_in[5];
    const float* alpha     = (const float*)d_in[6];
    const float* w2        = (const float*)d_in[7];
    const float* b2        = (const float*)d_in[8];
    const int*   edge_index = (const int*)d_in[9];

    const int n_nodes = in_sizes[0] / S_FEAT;
    const int n_edges = in_sizes[2] / E_FEAT;
    const int n_steps = in_sizes[3] / (D_FEAT * D_FEAT) - 1;   // K = 8

    const int* frm = edge_index;
    const int* to  = edge_index + n_edges;

    float* out = (float*)d_out;          // [N, 2]
    float* acc = (float*)d_ws;           // [N, 2] scratch for segment_sum

    const int nodeBlocks = (n_nodes + BLOCK - 1) / BLOCK;
    const int edgeBlocks = (n_edges + EDGES_PER_BLOCK - 1) / EDGES_PER_BLOCK;

    swegnn_init_kernel<<<nodeBlocks, BLOCK, 0, stream>>>(x_t, filt, out, acc, n_nodes);

    for (int k = 0; k < n_steps; ++k) {
        swegnn_edge_kernel<<<edgeBlocks, BLOCK, 0, stream>>>(
            x_s, out, edge_attr, w1, b1, alpha, w2, b2, frm, to, acc, n_edges);
        swegnn_node_kernel<<<nodeBlocks, BLOCK, 0, stream>>>(
            out, acc, filt + (size_t)(k + 1) * D_FEAT * D_FEAT, n_nodes);
    }
}